// net_26980984553961
// MI455X (gfx1250) — compile-verified
//
#include <hip/hip_runtime.h>
#include <hip/hip_bf16.h>
#include <math.h>

// Problem constants (match reference)
#define NN   50000
#define EE   100000
#define GG   2048
#define DD   32
#define FNODE 11
#define FEDGE 4
#define NCH  (FEDGE + 1)    // 4 edge-attr channels + bias channel
#define NTILES (EE / 16)    // 6250, exact

typedef __attribute__((ext_vector_type(2))) float v2f;
typedef __attribute__((ext_vector_type(8))) float v8f;

__device__ __forceinline__ float sigmoidf(float x) { return 1.0f / (1.0f + __expf(-x)); }

// ---------------------------------------------------------------------------
// Kernel 1: out = relu(x @ lin0_w.T + lin0_b); h = out
// ---------------------------------------------------------------------------
__global__ void k_lin0(const float* __restrict__ x, const float* __restrict__ w,
                       const float* __restrict__ b, float* __restrict__ out,
                       float* __restrict__ h) {
  int idx = blockIdx.x * blockDim.x + threadIdx.x;
  if (idx >= NN * DD) return;
  int n = idx >> 5, d = idx & 31;
  float acc = b[d];
  #pragma unroll
  for (int f = 0; f < FNODE; ++f) acc += x[n * FNODE + f] * w[d * FNODE + f];
  acc = fmaxf(acc, 0.0f);
  out[idx] = acc;
  h[idx] = acc;
}

// ---------------------------------------------------------------------------
// Kernel 2 (WMMA): per-edge message + scatter-add, W_e recomputed on the fly.
//   msg[e,:] = out[src[e],:] @ (sum_k ea[e,k]*M_k + M_bias)
// One wave per 16-edge tile. B fragments (tile-invariant) are packed in LDS
// as contiguous float2 so each V_WMMA_F32_16X16X4_F32 needs one ds_load_b64.
// Channel/half loops kept rolled to avoid register spills.
// ---------------------------------------------------------------------------
// s_B layout: index p = ((k*8 + c)*2 + hi)*32 + o  ->  float2 {M_k[i0,o], M_k[i0+1,o]}
// with i0 = 4*c + 2*hi;  M_k[i,o] = (k<4) ? nn_w[(i*32+o)*4+k] : nn_b[i*32+o]
#define NBPAIR (NCH * 8 * 2 * DD)   // 2560 float2 = 20 KB

__global__ void __launch_bounds__(256, 1)
k_msg(const float* __restrict__ out,
      const float* __restrict__ edge_attr,
      const float* __restrict__ nn_w,   // [1024,4]
      const float* __restrict__ nn_b,   // [1024]
      const int* __restrict__ src,
      const int* __restrict__ dst,
      float* __restrict__ aggr) {
  __shared__ float s_B[NBPAIR * 2];

  for (int p = threadIdx.x; p < NBPAIR; p += blockDim.x) {
    int k  = p >> 9;            // / 512
    int r  = p & 511;
    int c  = r >> 6;            // / 64
    int r2 = r & 63;
    int hi = r2 >> 5;
    int o  = r2 & 31;
    int i0 = 4 * c + 2 * hi;
    float f0, f1;
    if (k < FEDGE) {
      f0 = nn_w[(i0 * DD + o) * FEDGE + k];
      f1 = nn_w[((i0 + 1) * DD + o) * FEDGE + k];
    } else {
      f0 = nn_b[i0 * DD + o];
      f1 = nn_b[(i0 + 1) * DD + o];
    }
    s_B[2 * p]     = f0;
    s_B[2 * p + 1] = f1;
  }
  __syncthreads();

  const int lane = threadIdx.x & 31;
  const int wave = threadIdx.x >> 5;
  const int wavesPerBlock = blockDim.x >> 5;
  const int gwave = blockIdx.x * wavesPerBlock + wave;
  const int nwaves = gridDim.x * wavesPerBlock;

  const int row  = lane & 15;   // A row / C column (within 16)
  const int hi   = lane >> 4;   // lane half
  const int koff = hi * 2;      // K offset inside a K=4 chunk for this half

  for (int tile = gwave; tile < NTILES; tile += nwaves) {
    const int e0 = tile * 16;
    const int my_e = e0 + row;

    // Gather this row's node features (each lane holds half the K values)
    const float* orow = out + (long)src[my_e] * DD;
    v2f A[8];
    #pragma unroll
    for (int c = 0; c < 8; ++c)
      A[c] = *(const v2f*)(orow + 4 * c + koff);

    float ea[NCH];
    #pragma unroll
    for (int k = 0; k < FEDGE; ++k) ea[k] = edge_attr[my_e * FEDGE + k];
    ea[FEDGE] = 1.0f;   // bias channel

    #pragma unroll 1
    for (int hh = 0; hh < 2; ++hh) {
      const int o = hh * 16 + row;          // output column handled by this lane
      const float* bbase = s_B + 2 * ((size_t)0 + hi * 32 + o);  // + ((k*8+c)*64)*... below
      v8f acc = {};
      #pragma unroll 1
      for (int k = 0; k < NCH; ++k) {
        const float s = ea[k];
        #pragma unroll
        for (int c = 0; c < 8; ++c) {
          const int p = ((k * 8 + c) * 2 + hi) * DD + o;
          v2f bfrag = *(const v2f*)(s_B + 2 * p);
          v2f afrag;
          afrag.x = A[c].x * s;
          afrag.y = A[c].y * s;
          acc = __builtin_amdgcn_wmma_f32_16x16x4_f32(
              false, afrag, false, bfrag, (short)0, acc, false, false);
        }
      }
      (void)bbase;
      // Scatter-add: C layout — VGPR r holds edge row (r + 8*hi), column o
      #pragma unroll
      for (int r = 0; r < 8; ++r) {
        const int er = e0 + r + 8 * hi;
        atomicAdd(&aggr[(long)dst[er] * DD + o], acc[r]);
      }
    }
  }
}

// ---------------------------------------------------------------------------
// Kernel 3: node update (NNConv root + bias + relu) fused with GRU cell.
// One thread per node; weights broadcast-read through L2.
// ---------------------------------------------------------------------------
__global__ void k_node(const float* __restrict__ aggr,
                       const float* __restrict__ root_w,  // [32,32] (i,d)
                       const float* __restrict__ conv_b,
                       const float* __restrict__ wi,      // [96,32]
                       const float* __restrict__ wh,      // [96,32]
                       const float* __restrict__ bi,
                       const float* __restrict__ bh,
                       float* __restrict__ out_io,        // in: out, out: new h
                       float* __restrict__ h_io) {
  int n = blockIdx.x * blockDim.x + threadIdx.x;
  if (n >= NN) return;
  float o[DD], h[DD], m[DD];
  #pragma unroll
  for (int d = 0; d < DD; ++d) { o[d] = out_io[n * DD + d]; h[d] = h_io[n * DD + d]; }
  #pragma unroll 4
  for (int d = 0; d < DD; ++d) {
    float acc = aggr[n * DD + d] + conv_b[d];
    #pragma unroll
    for (int i = 0; i < DD; ++i) acc += o[i] * root_w[i * DD + d];
    m[d] = fmaxf(acc, 0.0f);
  }
  #pragma unroll 1
  for (int j = 0; j < DD; ++j) {
    float ir = bi[j],          hr = bh[j];
    float iz = bi[DD + j],     hz = bh[DD + j];
    float in_ = bi[2*DD + j],  hn = bh[2*DD + j];
    #pragma unroll
    for (int i = 0; i < DD; ++i) {
      ir  += m[i] * wi[j * DD + i];
      hr  += h[i] * wh[j * DD + i];
      iz  += m[i] * wi[(DD + j) * DD + i];
      hz  += h[i] * wh[(DD + j) * DD + i];
      in_ += m[i] * wi[(2*DD + j) * DD + i];
      hn  += h[i] * wh[(2*DD + j) * DD + i];
    }
    float r  = sigmoidf(ir + hr);
    float z  = sigmoidf(iz + hz);
    float nn = tanhf(in_ + r * hn);
    float hv = (1.0f - z) * nn + z * h[j];
    out_io[n * DD + j] = hv;
    h_io[n * DD + j]   = hv;
  }
}

// ---------------------------------------------------------------------------
// Kernel 4: mean-pool per graph (atomic sums + counts)
// ---------------------------------------------------------------------------
__global__ void k_pool(const float* __restrict__ out, const int* __restrict__ batch,
                       float* __restrict__ gsum, float* __restrict__ gcnt) {
  int idx = blockIdx.x * blockDim.x + threadIdx.x;
  if (idx >= NN * DD) return;
  int n = idx >> 5, d = idx & 31;
  int g = batch[n];
  atomicAdd(&gsum[g * DD + d], out[idx]);
  if (d == 0) atomicAdd(&gcnt[g], 1.0f);
}

// ---------------------------------------------------------------------------
// Kernel 5: logits + log_softmax
// ---------------------------------------------------------------------------
__global__ void k_logits(const float* __restrict__ gsum, const float* __restrict__ gcnt,
                         const float* __restrict__ w, const float* __restrict__ b,
                         float* __restrict__ outp) {
  int g = blockIdx.x * blockDim.x + threadIdx.x;
  if (g >= GG) return;
  float inv = 1.0f / fmaxf(gcnt[g], 1.0f);
  float l0 = b[0], l1 = b[1];
  #pragma unroll
  for (int d = 0; d < DD; ++d) {
    float p = gsum[g * DD + d] * inv;
    l0 += p * w[d];
    l1 += p * w[DD + d];
  }
  float mx = fmaxf(l0, l1);
  float lse = mx + logf(expf(l0 - mx) + expf(l1 - mx));
  outp[2 * g]     = l0 - lse;
  outp[2 * g + 1] = l1 - lse;
}

// ---------------------------------------------------------------------------
extern "C" void kernel_launch(void* const* d_in, const int* in_sizes, int n_in,
                              void* d_out, int out_size, void* d_ws, size_t ws_size,
                              hipStream_t stream) {
  (void)in_sizes; (void)n_in; (void)out_size; (void)ws_size;
  const float* x        = (const float*)d_in[0];
  const float* eattr    = (const float*)d_in[1];
  const float* lin0_w   = (const float*)d_in[2];
  const float* lin0_b   = (const float*)d_in[3];
  const float* nn_w     = (const float*)d_in[4];
  const float* nn_b     = (const float*)d_in[5];
  const float* root_w   = (const float*)d_in[6];
  const float* conv_b   = (const float*)d_in[7];
  const float* gru_wi   = (const float*)d_in[8];
  const float* gru_wh   = (const float*)d_in[9];
  const float* gru_bi   = (const float*)d_in[10];
  const float* gru_bh   = (const float*)d_in[11];
  const float* lin1_w   = (const float*)d_in[12];
  const float* lin1_b   = (const float*)d_in[13];
  const int*   eidx     = (const int*)d_in[14];
  const int*   batch    = (const int*)d_in[15];
  const int* src = eidx;
  const int* dst = eidx + EE;

  float* ws   = (float*)d_ws;
  float* out  = ws;                              // N*32
  float* h    = out  + (size_t)NN * DD;          // N*32
  float* aggr = h    + (size_t)NN * DD;          // N*32
  float* gsum = aggr + (size_t)NN * DD;          // G*32
  float* gcnt = gsum + (size_t)GG * DD;          // G

  // 1. input projection
  k_lin0<<<(NN * DD + 255) / 256, 256, 0, stream>>>(x, lin0_w, lin0_b, out, h);

  // 2. three message-passing + GRU steps
  for (int t = 0; t < 3; ++t) {
    hipMemsetAsync(aggr, 0, (size_t)NN * DD * sizeof(float), stream);
    k_msg<<<(NTILES + 7) / 8, 256, 0, stream>>>(out, eattr, nn_w, nn_b, src, dst, aggr);
    k_node<<<(NN + 127) / 128, 128, 0, stream>>>(aggr, root_w, conv_b,
                                                 gru_wi, gru_wh, gru_bi, gru_bh,
                                                 out, h);
  }

  // 3. mean-pool + classifier + log_softmax
  hipMemsetAsync(gsum, 0, (size_t)(GG * DD + GG) * sizeof(float), stream);
  k_pool<<<(NN * DD + 255) / 256, 256, 0, stream>>>(out, batch, gsum, gcnt);
  k_logits<<<(GG + 255) / 256, 256, 0, stream>>>(gsum, gcnt, lin1_w, lin1_b, (float*)d_out);
}